// SimpleGraphConv_56736517980586
// MI455X (gfx1250) — compile-verified
//
#include <hip/hip_runtime.h>

typedef __attribute__((ext_vector_type(2))) float v2f;
typedef __attribute__((ext_vector_type(8))) float v8f;

#define D_DIM 128
#define TILE 16

// -----------------------------------------------------------------------------
// Dual GEMM via V_WMMA_F32_16X16X4_F32 (wave32, one 16x16 tile per wave):
//   out  = x @ W_self^T + b_self
//   hnei = x @ W_nei^T
// A tile = x[m0:m0+16, k:k+4]   (16x4 f32, 2 VGPRs: lane holds K=2*(lane/16)+{0,1})
// B tile = W^T[k:k+4, n0:n0+16] (4x16 f32, 2 VGPRs: v0 holds K=0/K=2, v1 K=1/K=3,
//                                N = lane%16)  -> contiguous float2 from W's row.
// C/D: 8 VGPRs, row M = j + 8*(lane/16), col N = lane%16.
// -----------------------------------------------------------------------------
__global__ __launch_bounds__(256) void gc_dual_gemm(
    const float* __restrict__ x,
    const float* __restrict__ Wself,
    const float* __restrict__ bself,
    const float* __restrict__ Wnei,
    float* __restrict__ out,
    float* __restrict__ hnei,
    int N)
{
    const int lane   = threadIdx.x & 31;
    const int waveId = blockIdx.x * (blockDim.x >> 5) + (threadIdx.x >> 5);

    const int nTilesN = D_DIM / TILE;            // 8
    const int tileM = waveId / nTilesN;
    const int tileN = waveId % nTilesN;
    const int m0 = tileM * TILE;
    if (m0 >= N) return;                          // wave-uniform branch

    const int half = lane >> 4;                   // 0 or 1
    const int l16  = lane & 15;
    const int khi  = half * 2;                    // K sub-offset for A/B halves

    int arow = m0 + l16;
    if (arow >= N) arow = N - 1;                  // clamp (stores are guarded)
    const int bcol = tileN * TILE + l16;

    const float* __restrict__ xrow  = x     + (size_t)arow * D_DIM;
    const float* __restrict__ wsrow = Wself + (size_t)bcol * D_DIM;
    const float* __restrict__ wnrow = Wnei  + (size_t)bcol * D_DIM;

    v8f accS = {};
    v8f accN = {};
#pragma unroll
    for (int k = 0; k < D_DIM; k += 4) {
        v2f a  = *(const v2f*)(xrow  + k + khi);
        v2f bS = *(const v2f*)(wsrow + k + khi);
        v2f bN = *(const v2f*)(wnrow + k + khi);
        accS = __builtin_amdgcn_wmma_f32_16x16x4_f32(
                   false, a, false, bS, (short)0, accS, false, false);
        accN = __builtin_amdgcn_wmma_f32_16x16x4_f32(
                   false, a, false, bN, (short)0, accN, false, false);
    }

    const float bias  = bself[bcol];              // same column for all 8 elems
    const int   rbase = m0 + half * 8;
#pragma unroll
    for (int j = 0; j < 8; ++j) {
        const int row = rbase + j;
        if (row < N) {
            out [(size_t)row * D_DIM + bcol] = accS[j] + bias;
            hnei[(size_t)row * D_DIM + bcol] = accN[j];
        }
    }
}

// -----------------------------------------------------------------------------
// Edge scatter: one wave32 per edge. Each lane handles 4 consecutive floats
// (32*4 = 128). Gather h_nei[src] as float4, scale by edge weight, hardware
// fp32 atomic-add into out[dst] (lands in the 192MB L2 -> out fits entirely).
// -----------------------------------------------------------------------------
__global__ __launch_bounds__(256) void gc_edge_scatter(
    const long long* __restrict__ eidx,   // [2, E] int64
    const float* __restrict__ ew,         // [E]
    const float* __restrict__ hnei,       // [N, 128]
    float* __restrict__ out,              // [N, 128]
    int E)
{
    const int gid  = blockIdx.x * blockDim.x + threadIdx.x;
    const int e    = gid >> 5;                    // wave-uniform
    const int lane = threadIdx.x & 31;
    if (e >= E) return;

    const long long src = eidx[e];
    const long long dst = eidx[(size_t)E + e];
    const float     w   = ew[e];

    const float4 v = ((const float4*)(hnei + (size_t)src * D_DIM))[lane];
    float* op = out + (size_t)dst * D_DIM + lane * 4;

    unsafeAtomicAdd(op + 0, v.x * w);
    unsafeAtomicAdd(op + 1, v.y * w);
    unsafeAtomicAdd(op + 2, v.z * w);
    unsafeAtomicAdd(op + 3, v.w * w);
}

extern "C" void kernel_launch(void* const* d_in, const int* in_sizes, int n_in,
                              void* d_out, int out_size, void* d_ws, size_t ws_size,
                              hipStream_t stream) {
    const float*     x     = (const float*)d_in[0];
    const long long* eidx  = (const long long*)d_in[1];
    const float*     ew    = (const float*)d_in[2];
    const float*     Wself = (const float*)d_in[3];
    const float*     bself = (const float*)d_in[4];
    const float*     Wnei  = (const float*)d_in[5];
    float*           out   = (float*)d_out;
    float*           hnei  = (float*)d_ws;        // needs N*128*4 = 25.6 MB

    const int N = in_sizes[0] / D_DIM;
    const int E = in_sizes[2];

    // Phase 1: dual GEMM (writes every element of d_out, so no pre-zero needed)
    const int mTiles     = (N + TILE - 1) / TILE;
    const int wavesTotal = mTiles * (D_DIM / TILE);
    const int gblocks    = (wavesTotal + 7) / 8;  // 8 waves per 256-thread block
    gc_dual_gemm<<<gblocks, 256, 0, stream>>>(x, Wself, bself, Wnei, out, hnei, N);

    // Phase 2: edge gather/scale/scatter-add (same stream -> ordered after GEMM)
    const long long ethreads = (long long)E * 32;
    const int       eblocks  = (int)((ethreads + 255) / 256);
    gc_edge_scatter<<<eblocks, 256, 0, stream>>>(eidx, ew, hnei, out, E);
}